// Sampler_56599079027255
// MI455X (gfx1250) — compile-verified
//
#include <hip/hip_runtime.h>

#define BLOCK  512
#define NBINS  8192
#define BIN_SHIFT 19           // 32 - log2(NBINS)
#define CHUNK  (NBINS / BLOCK) // 16 bins per thread in the scan
#define CAP    512             // candidate capacity (== BLOCK for the bitonic sort)
#define NWAVES (BLOCK / 32)
#define TILE   8192            // floats per TDM tile (32 KB)

typedef __attribute__((ext_vector_type(4))) unsigned int u32x4;
typedef __attribute__((ext_vector_type(8))) int          i32x8;
typedef __attribute__((ext_vector_type(4))) int          i32x4;

#if defined(__has_builtin)
#  if __has_builtin(__builtin_amdgcn_tensor_load_to_lds)
#    define HAVE_TDM 1
#  endif
#endif
#ifndef HAVE_TDM
#  define HAVE_TDM 0
#endif

#if defined(__has_include)
#  if __has_include(<hip/amd_detail/amd_gfx1250_TDM.h>)
#    define TDM_SIX_ARG 1   // therock-10.0 headers => 6-arg builtin
#  else
#    define TDM_SIX_ARG 0   // ROCm 7.2 => 5-arg builtin
#  endif
#else
#  define TDM_SIX_ARG 0
#endif

// Monotone mapping float -> uint32 (order-preserving, including negatives)
__device__ __forceinline__ unsigned int f2key(float f) {
    unsigned int u = __float_as_uint(f);
    return (u & 0x80000000u) ? ~u : (u | 0x80000000u);
}
__device__ __forceinline__ float key2f(unsigned int k) {
    unsigned int u = (k & 0x80000000u) ? (k & 0x7fffffffu) : ~k;
    return __uint_as_float(u);
}

// wave32 reductions (CDNA5 is wave32-only)
__device__ __forceinline__ float waveSum(float v) {
#pragma unroll
    for (int o = 16; o > 0; o >>= 1) v += __shfl_down(v, o, 32);
    return v;
}
__device__ __forceinline__ float waveMax(float v) {
#pragma unroll
    for (int o = 16; o > 0; o >>= 1) v = fmaxf(v, __shfl_down(v, o, 32));
    return v;
}

#if HAVE_TDM
// Issue one 1-D TDM transfer: n_elems floats  global -> LDS byte offset.
// D# layout per CDNA5 ISA ch.8: group0 = {count/type/lds/global addr},
// group1 = {data_size, tensor_dim0, tile_dim0, strides}; groups 2/3 unused (zero).
__device__ __forceinline__ void tdm_load_1d(unsigned lds_byte_addr,
                                            unsigned long long gaddr,
                                            unsigned n_elems) {
    u32x4 g0;
    g0.x = 1u;                                                 // count=1 (valid user D#)
    g0.y = lds_byte_addr;                                      // lds_addr[31:0]
    g0.z = (unsigned)(gaddr & 0xffffffffull);                  // global_addr[31:0]
    g0.w = (unsigned)((gaddr >> 32) & 0x01ffffffull) | (2u << 30); // addr[56:32] | type=2

    i32x8 g1;
    g1[0] = (int)(2u << 16);                                   // data_size=2 (4 bytes)
    g1[1] = (int)((n_elems & 0xffffu) << 16);                  // tensor_dim0[15:0]
    g1[2] = (int)((n_elems >> 16) & 0xffffu);                  // tensor_dim0[31:16]
    g1[3] = (int)((n_elems & 0xffffu) << 16);                  // tile_dim0[15:0]
    g1[4] = 0;                                                 // tile_dim1/2 = 0 (1-D)
    g1[5] = (int)n_elems;                                      // tensor_dim0_stride (unused 1-D)
    g1[6] = 0;
    g1[7] = 0;

    i32x4 z4 = {0, 0, 0, 0};
#if TDM_SIX_ARG
    i32x8 z8 = {0, 0, 0, 0, 0, 0, 0, 0};
    __builtin_amdgcn_tensor_load_to_lds(g0, g1, z4, z4, z8, 0);
#else
    __builtin_amdgcn_tensor_load_to_lds(g0, g1, z4, z4, 0);
#endif
}
#endif

__global__ __launch_bounds__(BLOCK) void sampler_topk_kernel(
    const float* __restrict__ logits,
    const float* __restrict__ temperatures,
    const int*   __restrict__ top_ks,
    const float* __restrict__ top_ps,
    const float* __restrict__ min_ps,
    const float* __restrict__ top_n_sigmas,
    const float* __restrict__ uarr,
    const int*   __restrict__ need_min_p_ptr,
    int*         __restrict__ out,
    int V)
{
    __shared__ unsigned int        hist[NBINS];
    __shared__ unsigned long long  cand[CAP];
    __shared__ unsigned int        s_scan[BLOCK];
    __shared__ float s_rsum[NWAVES], s_rsq[NWAVES], s_rmx[NWAVES], s_rz[NWAVES];
    __shared__ float s_max, s_thr, s_Z;
    __shared__ int   s_bsel, s_ncand;
#if HAVE_TDM
    __shared__ float tileA[TILE], tileB[TILE];   // 2 x 32 KB double buffer
#endif

    const int row  = blockIdx.x;
    const int tid  = threadIdx.x;
    const int lane = tid & 31;
    const int wid  = tid >> 5;

    const float   temp  = temperatures[row];
    const float*  lrow  = logits + (size_t)row * (size_t)V;
    const int     V4    = V >> 2;
    const float4* lrow4 = (const float4*)lrow;

    for (int i = tid; i < NBINS; i += BLOCK) hist[i] = 0u;
    if (tid == 0) { s_ncand = 0; s_bsel = 0; }
    __syncthreads();

    // ---------- Pass 1 (HBM stream): sum / sumsq / max + key histogram ----------
    float lsum = 0.f, lsq = 0.f, lmx = -__builtin_inff();

#if HAVE_TDM
    // Tensor Data Mover streams 32 KB tiles into LDS; double-buffered so the DMA
    // of tile t+1 overlaps processing of tile t. One TDM issue per tile by wave 0.
    const int nTiles = (V + TILE - 1) / TILE;
    if (tid < 32) {   // wave 0 only (wave-uniform branch); TDM ignores EXEC
        int c0 = (V < TILE) ? V : TILE;
        tdm_load_1d((unsigned)(unsigned long long)(void*)tileA,
                    (unsigned long long)(size_t)(const void*)lrow, (unsigned)c0);
    }
    for (int t = 0; t < nTiles; ++t) {
        float* cur = (t & 1) ? tileB : tileA;
        int    rem = V - t * TILE;
        int    cnt = (rem < TILE) ? rem : TILE;
        if (tid < 32) {
            int tn = t + 1;
            if (tn < nTiles) {
                float* nxt  = (tn & 1) ? tileB : tileA;
                int    remn = V - tn * TILE;
                int    cn   = (remn < TILE) ? remn : TILE;
                tdm_load_1d((unsigned)(unsigned long long)(void*)nxt,
                            (unsigned long long)(size_t)(const void*)(lrow + (size_t)tn * TILE),
                            (unsigned)cn);
                __builtin_amdgcn_s_wait_tensorcnt(1);  // tile t done, t+1 in flight
            } else {
                __builtin_amdgcn_s_wait_tensorcnt(0);
            }
        }
        __syncthreads();   // publish tile t to all 16 waves
        const int     cnt4 = cnt >> 2;
        const float4* c4   = (const float4*)cur;
        const int     gbase = t * TILE;
        for (int i = tid; i < cnt4; i += BLOCK) {
            float4 x = c4[i];
            float a = x.x / temp, b = x.y / temp, c = x.z / temp, d = x.w / temp;
            lsum += (a + b) + (c + d);
            lsq  += (a * a + b * b) + (c * c + d * d);
            lmx   = fmaxf(fmaxf(lmx, a), fmaxf(b, fmaxf(c, d)));
            atomicAdd(&hist[f2key(a) >> BIN_SHIFT], 1u);
            atomicAdd(&hist[f2key(b) >> BIN_SHIFT], 1u);
            atomicAdd(&hist[f2key(c) >> BIN_SHIFT], 1u);
            atomicAdd(&hist[f2key(d) >> BIN_SHIFT], 1u);
        }
        for (int i = (cnt4 << 2) + tid; i < cnt; i += BLOCK) {
            float a = cur[i] / temp;
            lsum += a; lsq += a * a; lmx = fmaxf(lmx, a);
            atomicAdd(&hist[f2key(a) >> BIN_SHIFT], 1u);
        }
        (void)gbase;
        __syncthreads();   // tile buffer reusable for t+2
    }
#else
    for (int i = tid; i < V4; i += BLOCK) {
        int pf = i + BLOCK * 4;
        if (pf < V4) __builtin_prefetch((const void*)(lrow4 + pf), 0, 0);
        float4 x = lrow4[i];
        float a = x.x / temp, b = x.y / temp, c = x.z / temp, d = x.w / temp;
        lsum += (a + b) + (c + d);
        lsq  += (a * a + b * b) + (c * c + d * d);
        lmx   = fmaxf(fmaxf(lmx, a), fmaxf(b, fmaxf(c, d)));
        atomicAdd(&hist[f2key(a) >> BIN_SHIFT], 1u);
        atomicAdd(&hist[f2key(b) >> BIN_SHIFT], 1u);
        atomicAdd(&hist[f2key(c) >> BIN_SHIFT], 1u);
        atomicAdd(&hist[f2key(d) >> BIN_SHIFT], 1u);
    }
    for (int i = (V4 << 2) + tid; i < V; i += BLOCK) {
        float a = lrow[i] / temp;
        lsum += a; lsq += a * a; lmx = fmaxf(lmx, a);
        atomicAdd(&hist[f2key(a) >> BIN_SHIFT], 1u);
    }
#endif

    float ws = waveSum(lsum), wq = waveSum(lsq), wm = waveMax(lmx);
    if (lane == 0) { s_rsum[wid] = ws; s_rsq[wid] = wq; s_rmx[wid] = wm; }
    __syncthreads();
    if (tid == 0) {
        float S = 0.f, Q = 0.f, M = -__builtin_inff();
        for (int i = 0; i < NWAVES; ++i) { S += s_rsum[i]; Q += s_rsq[i]; M = fmaxf(M, s_rmx[i]); }
        float N   = (float)V;
        float var = (Q - S * S / N) / (N - 1.0f);   // ddof=1
        float sg  = sqrtf(fmaxf(var, 0.f));
        float tns = top_n_sigmas[row];
        s_max = M;
        s_thr = (tns > 0.f) ? (M - tns * sg) : -__builtin_inff();
    }
    __syncthreads();

    // ---------- Find boundary bin of the 64th-largest key (parallel scan from top) ----------
    {
        const unsigned int K = (V < 64) ? (unsigned int)V : 64u;
        int base = NBINS - 1 - tid * CHUNK;           // chunks ordered from top bin down
        unsigned int csum = 0;
        for (int t2 = 0; t2 < CHUNK; ++t2) csum += hist[base - t2];
        s_scan[tid] = csum;
        __syncthreads();
        for (int off = 1; off < BLOCK; off <<= 1) {   // Hillis-Steele inclusive scan
            unsigned int v   = s_scan[tid];
            unsigned int add = (tid >= off) ? s_scan[tid - off] : 0u;
            __syncthreads();
            s_scan[tid] = v + add;
            __syncthreads();
        }
        unsigned int incl = s_scan[tid];
        unsigned int excl = incl - csum;
        if (incl >= K && excl < K) {                  // exactly one thread owns the crossing
            unsigned int acc = excl;
            for (int t2 = 0; t2 < CHUNK; ++t2) {
                acc += hist[base - t2];
                if (acc >= K) { s_bsel = base - t2; break; }
            }
        }
        __syncthreads();
    }

    const float thr  = s_thr;
    const float mx   = s_max;
    const int   bsel = s_bsel;

    // ---------- Pass 2 (L2-resident): softmax denominator + candidate collection ----------
    float lz = 0.f;
#define PROC(vv, ii) do {                                                            \
        if ((vv) >= thr) {                                                           \
            lz += __expf((vv) - mx);                                                 \
            unsigned int k_ = f2key(vv);                                             \
            if ((int)(k_ >> BIN_SHIFT) >= bsel) {                                    \
                int slot = atomicAdd(&s_ncand, 1);                                   \
                if (slot < CAP)                                                      \
                    cand[slot] = ((unsigned long long)k_ << 32) |                    \
                                 (unsigned long long)(~(unsigned int)(ii));          \
            }                                                                        \
        }                                                                            \
    } while (0)

    for (int i = tid; i < V4; i += BLOCK) {
        int pf = i + BLOCK * 4;
        if (pf < V4) __builtin_prefetch((const void*)(lrow4 + pf), 0, 0); // global_prefetch_b8
        float4 x = lrow4[i];
        float v0 = x.x / temp, v1 = x.y / temp, v2 = x.z / temp, v3 = x.w / temp;
        int bidx = i << 2;
        PROC(v0, bidx + 0);
        PROC(v1, bidx + 1);
        PROC(v2, bidx + 2);
        PROC(v3, bidx + 3);
    }
    for (int i = (V4 << 2) + tid; i < V; i += BLOCK) {
        float vv = lrow[i] / temp;
        PROC(vv, i);
    }
#undef PROC

    float wz = waveSum(lz);
    if (lane == 0) s_rz[wid] = wz;
    __syncthreads();
    if (tid == 0) {
        float Zt = 0.f;
        for (int i = 0; i < NWAVES; ++i) Zt += s_rz[i];
        s_Z = Zt;
    }
    __syncthreads();

    // ---------- Pad + bitonic sort (descending); tie-break = ascending index via ~idx ----------
    int n = s_ncand; if (n > CAP) n = CAP;
    for (int i = tid; i < CAP; i += BLOCK) if (i >= n) cand[i] = 0ull; // 0 == minimal key
    for (int k = 2; k <= CAP; k <<= 1) {
        for (int j = k >> 1; j > 0; j >>= 1) {
            __syncthreads();
            int i   = tid;          // BLOCK == CAP: one pair per thread per round
            int ixj = i ^ j;
            if (ixj > i) {
                unsigned long long a = cand[i], b = cand[ixj];
                bool desc = ((i & k) == 0);
                if (desc ? (a < b) : (a > b)) { cand[i] = b; cand[ixj] = a; }
            }
        }
    }
    __syncthreads();

    // ---------- Sequential top-k / top-p / min-p masking + inverse-CDF sample ----------
    if (tid == 0) {
        const int   need_min_p = need_min_p_ptr[0];
        const int   tk = top_ks[row];
        const float tp = top_ps[row];
        const float mp = min_ps[row];
        const float uu = uarr[row];
        const float Z  = s_Z;

        int lim = (n < 64) ? n : 64;
        if (lim < 1) lim = 1;

        float p0 = __expf(key2f((unsigned int)(cand[0] >> 32)) - mx) / Z;

        // loop 1: total surviving mass (mask uses EXCLUSIVE prefix of unmasked ps, like csum - ps)
        float excl = 0.f, total = 0.f;
        for (int i = 0; i < lim; ++i) {
            float li = key2f((unsigned int)(cand[i] >> 32));
            float p  = __expf(li - mx) / Z;
            bool  msk = (i >= tk) || (excl > tp);
            excl += p;
            float p2 = msk ? 0.f : p;
            if (need_min_p && (p2 < p0 * mp)) p2 = 0.f;
            total += p2;
        }
        float r = uu * total;

        // loop 2: recompute identical sequence, count cum < r
        excl = 0.f;
        float cum = 0.f;
        int   cnt = 0;
        for (int i = 0; i < lim; ++i) {
            float li = key2f((unsigned int)(cand[i] >> 32));
            float p  = __expf(li - mx) / Z;
            bool  msk = (i >= tk) || (excl > tp);
            excl += p;
            float p2 = msk ? 0.f : p;
            if (need_min_p && (p2 < p0 * mp)) p2 = 0.f;
            cum += p2;
            if (cum < r) ++cnt;
        }
        if (cnt > lim - 1) cnt = lim - 1;

        out[row] = (int)(~(unsigned int)(cand[cnt] & 0xffffffffull));
    }
}

extern "C" void kernel_launch(void* const* d_in, const int* in_sizes, int n_in,
                              void* d_out, int out_size, void* d_ws, size_t ws_size,
                              hipStream_t stream) {
    const float* logits = (const float*)d_in[0];
    const float* temps  = (const float*)d_in[1];
    const int*   topks  = (const int*)  d_in[2];
    const float* topps  = (const float*)d_in[3];
    const float* minps  = (const float*)d_in[4];
    const float* tns    = (const float*)d_in[5];
    const float* u      = (const float*)d_in[6];
    const int*   nmp    = (const int*)  d_in[7];

    const int B = in_sizes[1];             // temperatures has B elements
    const int V = in_sizes[0] / B;

    sampler_topk_kernel<<<B, BLOCK, 0, stream>>>(
        logits, temps, topks, topps, minps, tns, u, nmp, (int*)d_out, V);
}